// InsSegTest_86457691668808
// MI455X (gfx1250) — compile-verified
//
#include <hip/hip_runtime.h>
#include <hip/hip_bf16.h>

typedef __attribute__((ext_vector_type(2))) float v2f;
typedef __attribute__((ext_vector_type(8))) float v8f;

// ---------------- static problem config (mirrors reference) ----------------
constexpr int cB = 4, cN = 100000, cG = 64, cP = 16;
constexpr int NPTS   = cB * cN;            // 400000
constexpr int NVOX   = cB * cG * cG * cG;  // 1048576
constexpr int NPR    = cB * cP * cP * cP;  // 16384
constexpr int NTILES = NPTS / 16;          // 25000 (exactly; 16 | 100000)
constexpr int TPBATCH = cN / 16;           // 6250 tiles per batch
constexpr float VOXSZ = 0.01f, RADIUS = 0.12f, BN_EPS = 1e-4f;

// ---------------- workspace layout (bytes) ----------------
constexpr size_t alignup(size_t x) { return (x + 255) & ~(size_t)255; }
constexpr size_t O_VSUM  = 0;                                        // NVOX*6 f32
constexpr size_t O_CNT   = alignup(O_VSUM  + (size_t)NVOX * 6 * 4);  // NVOX f32
constexpr size_t O_VID   = alignup(O_CNT   + (size_t)NVOX * 4);      // NPTS i32
constexpr size_t O_PC    = alignup(O_VID   + (size_t)NPTS * 4);      // NPTS*16 f32
constexpr size_t O_PREDS = alignup(O_PC    + (size_t)NPTS * 16 * 4); // NPTS i32
constexpr size_t O_PID   = alignup(O_PREDS + (size_t)NPTS * 4);      // NPTS i32
constexpr size_t O_POOL  = alignup(O_PID   + (size_t)NPTS * 4);      // NPR*16 u32 (float bits)
constexpr size_t O_PROP  = alignup(O_POOL  + (size_t)NPR * 16 * 4);  // NPR i32
constexpr size_t O_STATS = alignup(O_PROP  + (size_t)NPR * 4);       // 32 f64
constexpr size_t O_BN    = alignup(O_STATS + 32 * 8);                // 32 f32 (scale,shift)
constexpr size_t O_VMINK = alignup(O_BN    + 32 * 4);                // 12 u32
constexpr size_t O_SMINK = alignup(O_VMINK + 12 * 4);                // 12 u32
constexpr size_t WS_BYTES = alignup(O_SMINK + 12 * 4);               // ~61 MB

// ---------------- helpers ----------------
__device__ __forceinline__ unsigned fkey(float f) {   // order-preserving uint key
  unsigned b = __float_as_uint(f);
  return (b & 0x80000000u) ? ~b : (b | 0x80000000u);
}
__device__ __forceinline__ float funkey(unsigned k) {
  unsigned b = (k & 0x80000000u) ? (k & 0x7FFFFFFFu) : ~k;
  return __uint_as_float(b);
}
__device__ __forceinline__ unsigned umin2(unsigned a, unsigned b) { return a < b ? a : b; }

// One 16(M) x 16(N) x 16(K) f32 tile GEMM as 4 chained V_WMMA_F32_16X16X4_F32.
// A: row-major [16 x K=16] with leading dim lda (global or LDS).
// Bw: row-major [16 x bcols]; logical N columns are [noff, noff+16), padded with 0.
// Padding is branchless (clamped index * 0/1 mask) so the compiler emits
// v_cndmask + plain loads instead of exec-mask branches around the B loads,
// keeping the WMMA issue stream tight.
// A layout per ISA: VGPR0 = K{0|2}, VGPR1 = K{1|3} (lanes 0-15 | 16-31), M = lane&15.
// B layout (rows striped across lanes per VGPR): VGPR0 = K{0|2}, VGPR1 = K{1|3}, N = lane&15.
__device__ __forceinline__ v8f wmma_tile16(const float* A, int lda,
                                           const float* Bw, int bcols, int noff,
                                           int lane) {
  const int half = lane >> 4;
  const int l = lane & 15;
  const int n = noff + l;
  const int nn = (n < bcols) ? n : 0;          // clamped column
  const float msk = (n < bcols) ? 1.f : 0.f;   // zero-pad mask
  v8f c = {0.f, 0.f, 0.f, 0.f, 0.f, 0.f, 0.f, 0.f};
#pragma unroll
  for (int ko = 0; ko < 4; ++ko) {
    const int k0 = 4 * ko + 2 * half;
    v2f a, b;
    a.x = A[l * lda + k0];
    a.y = A[l * lda + k0 + 1];
    b.x = Bw[k0 * bcols + nn] * msk;
    b.y = Bw[(k0 + 1) * bcols + nn] * msk;
    c = __builtin_amdgcn_wmma_f32_16x16x4_f32(false, a, false, b, (short)0, c,
                                              false, false);
  }
  return c;
}

// ---------------- kernels ----------------
__global__ void k_zero(unsigned* p, size_t nwords) {
  for (size_t i = (size_t)blockIdx.x * blockDim.x + threadIdx.x; i < nwords;
       i += (size_t)gridDim.x * blockDim.x)
    p[i] = 0u;
}

__global__ void k_init_keys(unsigned* vmink, unsigned* smink) {
  int t = threadIdx.x;
  if (t < 12) vmink[t] = 0xFFFFFFFFu;
  else if (t < 24) smink[t - 12] = 0xFFFFFFFFu;
}

__global__ void k_vmin(const float* __restrict__ pts, unsigned* vmink) {
  const int b = blockIdx.y;
  float m0 = 3.4e38f, m1 = 3.4e38f, m2 = 3.4e38f;
  for (int i = blockIdx.x * blockDim.x + threadIdx.x; i < cN;
       i += gridDim.x * blockDim.x) {
    const float* p = pts + ((size_t)b * cN + i) * 3;
    m0 = fminf(m0, p[0]); m1 = fminf(m1, p[1]); m2 = fminf(m2, p[2]);
  }
  unsigned k0 = fkey(m0), k1 = fkey(m1), k2 = fkey(m2);
#pragma unroll
  for (int off = 16; off >= 1; off >>= 1) {
    k0 = umin2(k0, (unsigned)__shfl_xor((int)k0, off, 32));
    k1 = umin2(k1, (unsigned)__shfl_xor((int)k1, off, 32));
    k2 = umin2(k2, (unsigned)__shfl_xor((int)k2, off, 32));
  }
  if ((threadIdx.x & 31) == 0) {
    atomicMin(&vmink[b * 3 + 0], k0);
    atomicMin(&vmink[b * 3 + 1], k1);
    atomicMin(&vmink[b * 3 + 2], k2);
  }
}

__global__ void k_voxelize(const float* __restrict__ pts,
                           const float* __restrict__ flow,
                           const unsigned* __restrict__ vmink,
                           float* vsum, float* cnt, int* vid) {
  const int p = blockIdx.x * blockDim.x + threadIdx.x;
  if (p >= NPTS) return;
  const int b = p / cN, n = p - b * cN;
  int cc[3];
#pragma unroll
  for (int j = 0; j < 3; ++j) {
    const float vm = funkey(vmink[b * 3 + j]) - 1e-4f;
    int ci = (int)floorf((pts[(size_t)p * 3 + j] - vm) / VOXSZ);
    cc[j] = ci < 0 ? 0 : (ci > cG - 1 ? cG - 1 : ci);
  }
  const int v = ((b * cG + cc[0]) * cG + cc[1]) * cG + cc[2];
  vid[p] = v;
#pragma unroll
  for (int j = 0; j < 3; ++j) {
    atomicAdd(&vsum[(size_t)v * 6 + j], pts[(size_t)p * 3 + j]);
    atomicAdd(&vsum[(size_t)v * 6 + 3 + j], flow[((size_t)b * 3 + j) * cN + n]);
  }
  atomicAdd(&cnt[v], 1.0f);
}

__global__ void k_pointfeat(const float* __restrict__ vsum,
                            const float* __restrict__ cnt,
                            const int* __restrict__ vid,
                            const float* __restrict__ W1, const float* __restrict__ b1,
                            const float* __restrict__ W2, const float* __restrict__ b2,
                            float* pc) {
  const int p = blockIdx.x * blockDim.x + threadIdx.x;
  if (p >= NPTS) return;
  const int v = vid[p];
  const float inv = 1.0f / fmaxf(cnt[v], 1.0f);
  float f[6];
#pragma unroll
  for (int j = 0; j < 6; ++j) f[j] = vsum[(size_t)v * 6 + j] * inv;
  float h[16];
#pragma unroll
  for (int c = 0; c < 16; ++c) {
    float s = b1[c];
#pragma unroll
    for (int j = 0; j < 6; ++j) s += f[j] * W1[j * 16 + c];
    h[c] = fmaxf(s, 0.f);
  }
#pragma unroll
  for (int c = 0; c < 16; ++c) {
    float s = b2[c];
#pragma unroll
    for (int k = 0; k < 16; ++k) s += h[k] * W2[k * 16 + c];
    pc[(size_t)p * 16 + c] = s;
  }
}

__global__ void k_bnstats(const float* __restrict__ pc,
                          const float* __restrict__ Woff1,
                          const float* __restrict__ boff1, double* stats) {
  const int p = blockIdx.x * blockDim.x + threadIdx.x;
  const bool active = (p < NPTS);
  float a[16];
#pragma unroll
  for (int k = 0; k < 16; ++k) a[k] = active ? pc[(size_t)p * 16 + k] : 0.f;
#pragma unroll
  for (int c = 0; c < 16; ++c) {
    float o = active ? boff1[c] : 0.f;
#pragma unroll
    for (int k = 0; k < 16; ++k) o += a[k] * Woff1[k * 16 + c];
    float ws = o, wq = o * o;
#pragma unroll
    for (int off = 16; off >= 1; off >>= 1) {
      ws += __shfl_xor(ws, off, 32);
      wq += __shfl_xor(wq, off, 32);
    }
    if ((threadIdx.x & 31) == 0) {
      atomicAdd(&stats[c], (double)ws);
      atomicAdd(&stats[16 + c], (double)wq);
    }
  }
}

__global__ void k_bnfinal(const double* __restrict__ stats,
                          const float* __restrict__ gamma,
                          const float* __restrict__ beta, float* bn) {
  const int c = threadIdx.x;
  if (c < 16) {
    const double mean = stats[c] / (double)NPTS;
    const double var = stats[16 + c] / (double)NPTS - mean * mean;
    const float s = gamma[c] * rsqrtf((float)var + BN_EPS);
    bn[c] = s;
    bn[16 + c] = beta[c] - (float)mean * s;
  }
}

__global__ void __launch_bounds__(256)
k_heads(const float* __restrict__ pc, const float* __restrict__ pts,
        const float* __restrict__ Wsem, const float* __restrict__ bsem,
        const float* __restrict__ Woff1, const float* __restrict__ boff1,
        const float* __restrict__ bn,
        const float* __restrict__ Woff2, const float* __restrict__ boff2,
        const float* __restrict__ Wnu, const float* __restrict__ bnu,
        const float* __restrict__ Wnpcs, const float* __restrict__ bnpcs,
        float* out_sem, float* out_off, float* out_npcs, unsigned* smink) {
  __shared__ float lds[8][16][17];  // per-wave 16x16 transpose tile (+1 pad)
  const int wave = threadIdx.x >> 5, lane = threadIdx.x & 31;
  const int tile = blockIdx.x * 8 + wave;
  const int pt0 = tile * 16;
  const int b = tile / TPBATCH;  // tiles never straddle batches (16 | 100000)
  const int half = lane >> 4, l = lane & 15;
  const float* A = pc + (size_t)pt0 * 16;

  // ---- semantic head: [16x16] @ [16x10] ----
  {
    v8f cs = wmma_tile16(A, 16, Wsem, 10, 0, lane);
    if (l < 10) {
      const float bias = bsem[l];
#pragma unroll
      for (int r = 0; r < 8; ++r)
        out_sem[(size_t)(pt0 + r + 8 * half) * 10 + l] = cs[r] + bias;
    }
  }

  // ---- offset head: WMMA -> BN+ReLU -> LDS transpose -> WMMA [16x3] ----
  {
    v8f o = wmma_tile16(A, 16, Woff1, 16, 0, lane);
    const float s = bn[l], t = bn[16 + l], bo = boff1[l];
#pragma unroll
    for (int r = 0; r < 8; ++r)
      lds[wave][r + 8 * half][l] = fmaxf((o[r] + bo) * s + t, 0.f);
  }
  __syncthreads();
  {
    v8f d2 = wmma_tile16(&lds[wave][0][0], 17, Woff2, 3, 0, lane);
    unsigned mk = 0xFFFFFFFFu;
    if (l < 3) {
      const float bias = boff2[l];
#pragma unroll
      for (int r = 0; r < 8; ++r) {
        const int p = pt0 + r + 8 * half;
        const float off = d2[r] + bias;
        out_off[(size_t)p * 3 + l] = off;
        mk = umin2(mk, fkey(pts[(size_t)p * 3 + l] + off));  // shifted coord
      }
    }
    mk = umin2(mk, (unsigned)__shfl_xor((int)mk, 16, 32));
    if (lane < 3) atomicMin(&smink[b * 3 + lane], mk);
  }
  __syncthreads();

  // ---- npcs head: WMMA -> ReLU -> LDS transpose -> 2 WMMA column tiles ----
  {
    v8f nf = wmma_tile16(A, 16, Wnu, 16, 0, lane);
    const float bias = bnu[l];
#pragma unroll
    for (int r = 0; r < 8; ++r)
      lds[wave][r + 8 * half][l] = fmaxf(nf[r] + bias, 0.f);
  }
  __syncthreads();
  {
    v8f n0 = wmma_tile16(&lds[wave][0][0], 17, Wnpcs, 27, 0, lane);
    v8f n1 = wmma_tile16(&lds[wave][0][0], 17, Wnpcs, 27, 16, lane);
    const float bias0 = bnpcs[l];
#pragma unroll
    for (int r = 0; r < 8; ++r)
      out_npcs[(size_t)(pt0 + r + 8 * half) * 27 + l] = n0[r] + bias0;
    if (l < 11) {
      const float bias1 = bnpcs[16 + l];
#pragma unroll
      for (int r = 0; r < 8; ++r)
        out_npcs[(size_t)(pt0 + r + 8 * half) * 27 + 16 + l] = n1[r] + bias1;
    }
  }
}

__global__ void k_argmax(const float* __restrict__ out_sem, int* preds) {
  const int p = blockIdx.x * blockDim.x + threadIdx.x;
  if (p >= NPTS) return;
  const float* s = out_sem + (size_t)p * 10;
  float best = s[0];
  int bi = 0;
#pragma unroll
  for (int c = 1; c < 10; ++c)
    if (s[c] > best) { best = s[c]; bi = c; }  // first max, like jnp.argmax
  preds[p] = bi;
}

__global__ void k_pid(const float* __restrict__ pts,
                      const float* __restrict__ out_off,
                      const unsigned* __restrict__ smink, int* pid) {
  const int p = blockIdx.x * blockDim.x + threadIdx.x;
  if (p >= NPTS) return;
  const int b = p / cN;
  int cc[3];
#pragma unroll
  for (int j = 0; j < 3; ++j) {
    const float sm = funkey(smink[b * 3 + j]) - 1e-4f;
    const float sh = pts[(size_t)p * 3 + j] + out_off[(size_t)p * 3 + j];
    int ci = (int)floorf((sh - sm) / RADIUS);
    cc[j] = ci < 0 ? 0 : (ci > cP - 1 ? cP - 1 : ci);
  }
  pid[p] = ((b * cP + cc[0]) * cP + cc[1]) * cP + cc[2];
}

__global__ void __launch_bounds__(256)
k_pool(const float* __restrict__ pc, const float* __restrict__ Wsu,
       const float* __restrict__ bsu, const int* __restrict__ preds,
       const int* __restrict__ pid, unsigned* pooled, int* prop) {
  const int wave = threadIdx.x >> 5, lane = threadIdx.x & 31;
  const int tile = blockIdx.x * 8 + wave;
  const int pt0 = tile * 16;
  const int half = lane >> 4, l = lane & 15;
  v8f sf = wmma_tile16(pc + (size_t)pt0 * 16, 16, Wsu, 16, 0, lane);
  const float bias = bsu[l];
#pragma unroll
  for (int r = 0; r < 8; ++r) {
    const int p = pt0 + r + 8 * half;
    const int sp = preds[p];
    if (sp > 0) {  // foreground only; ReLU output >= 0 so raw-bit uint max == float max
      const float v = fmaxf(sf[r] + bias, 0.f);
      atomicMax(&pooled[(size_t)pid[p] * 16 + l], __float_as_uint(v));
      if (l == 0) atomicMax(&prop[pid[p]], sp);
    }
  }
}

__global__ void k_score(const unsigned* __restrict__ pooled,
                        const int* __restrict__ prop,
                        const float* __restrict__ Wscore,
                        const float* __restrict__ bscore, float* out_score) {
  const int pr = blockIdx.x * blockDim.x + threadIdx.x;
  if (pr >= NPR) return;
  int cls = prop[pr] - 1;
  cls = cls < 0 ? 0 : (cls > 8 ? 8 : cls);
  float s = bscore[cls];
#pragma unroll
  for (int k = 0; k < 16; ++k)
    s += __uint_as_float(pooled[(size_t)pr * 16 + k]) * Wscore[k * 9 + cls];
  out_score[pr] = s;
}

// ---------------- launch ----------------
extern "C" void kernel_launch(void* const* d_in, const int* in_sizes, int n_in,
                              void* d_out, int out_size, void* d_ws, size_t ws_size,
                              hipStream_t stream) {
  (void)in_sizes; (void)n_in; (void)out_size;
  if (ws_size < WS_BYTES) return;

  const float* pts   = (const float*)d_in[0];
  const float* flow  = (const float*)d_in[1];
  const float* W1    = (const float*)d_in[2];
  const float* b1    = (const float*)d_in[3];
  const float* W2    = (const float*)d_in[4];
  const float* b2    = (const float*)d_in[5];
  const float* Wsem  = (const float*)d_in[6];
  const float* bsem  = (const float*)d_in[7];
  const float* Woff1 = (const float*)d_in[8];
  const float* boff1 = (const float*)d_in[9];
  const float* gamma = (const float*)d_in[10];
  const float* beta  = (const float*)d_in[11];
  const float* Woff2 = (const float*)d_in[12];
  const float* boff2 = (const float*)d_in[13];
  const float* Wsu   = (const float*)d_in[14];
  const float* bsu   = (const float*)d_in[15];
  const float* Wscor = (const float*)d_in[16];
  const float* bscor = (const float*)d_in[17];
  const float* Wnu   = (const float*)d_in[18];
  const float* bnu   = (const float*)d_in[19];
  const float* Wnpcs = (const float*)d_in[20];
  const float* bnpcs = (const float*)d_in[21];

  char* ws = (char*)d_ws;
  float*    vsum  = (float*)(ws + O_VSUM);
  float*    cnt   = (float*)(ws + O_CNT);
  int*      vid   = (int*)(ws + O_VID);
  float*    pc    = (float*)(ws + O_PC);
  int*      preds = (int*)(ws + O_PREDS);
  int*      pid   = (int*)(ws + O_PID);
  unsigned* pooled= (unsigned*)(ws + O_POOL);
  int*      prop  = (int*)(ws + O_PROP);
  double*   stats = (double*)(ws + O_STATS);
  float*    bn    = (float*)(ws + O_BN);
  unsigned* vmink = (unsigned*)(ws + O_VMINK);
  unsigned* smink = (unsigned*)(ws + O_SMINK);

  float* out = (float*)d_out;
  float* out_sem   = out;
  float* out_off   = out_sem + (size_t)NPTS * 10;
  float* out_score = out_off + (size_t)NPTS * 3;
  float* out_npcs  = out_score + NPR;

  const int ptBlocks = (NPTS + 255) / 256;   // 1563
  const int tileBlocks = NTILES / 8;         // 3125 (exact)

  hipLaunchKernelGGL(k_zero, dim3(2048), dim3(256), 0, stream,
                     (unsigned*)d_ws, WS_BYTES / 4);
  hipLaunchKernelGGL(k_init_keys, dim3(1), dim3(32), 0, stream, vmink, smink);
  hipLaunchKernelGGL(k_vmin, dim3(64, cB), dim3(256), 0, stream, pts, vmink);
  hipLaunchKernelGGL(k_voxelize, dim3(ptBlocks), dim3(256), 0, stream,
                     pts, flow, vmink, vsum, cnt, vid);
  hipLaunchKernelGGL(k_pointfeat, dim3(ptBlocks), dim3(256), 0, stream,
                     vsum, cnt, vid, W1, b1, W2, b2, pc);
  hipLaunchKernelGGL(k_bnstats, dim3(ptBlocks), dim3(256), 0, stream,
                     pc, Woff1, boff1, stats);
  hipLaunchKernelGGL(k_bnfinal, dim3(1), dim3(16), 0, stream,
                     stats, gamma, beta, bn);
  hipLaunchKernelGGL(k_heads, dim3(tileBlocks), dim3(256), 0, stream,
                     pc, pts, Wsem, bsem, Woff1, boff1, bn, Woff2, boff2,
                     Wnu, bnu, Wnpcs, bnpcs, out_sem, out_off, out_npcs, smink);
  hipLaunchKernelGGL(k_argmax, dim3(ptBlocks), dim3(256), 0, stream,
                     out_sem, preds);
  hipLaunchKernelGGL(k_pid, dim3(ptBlocks), dim3(256), 0, stream,
                     pts, out_off, smink, pid);
  hipLaunchKernelGGL(k_pool, dim3(tileBlocks), dim3(256), 0, stream,
                     pc, Wsu, bsu, preds, pid, pooled, prop);
  hipLaunchKernelGGL(k_score, dim3((NPR + 255) / 256), dim3(256), 0, stream,
                     pooled, prop, Wscor, bscor, out_score);
}